// AdaptiveEmbedding_84499186581597
// MI455X (gfx1250) — compile-verified
//
#include <hip/hip_runtime.h>
#include <hip/hip_bf16.h>

typedef float v2f __attribute__((ext_vector_type(2)));
typedef float v4f __attribute__((ext_vector_type(4)));
typedef float v8f __attribute__((ext_vector_type(8)));

#define D_PROJ   1024
#define EMB_SCALE 32.0f   // sqrt(1024)

// ---------------- pass 0: zero the 4 cluster counters ----------------
__global__ void init_counters(int* __restrict__ cnt) {
    if (threadIdx.x < 4) cnt[threadIdx.x] = 0;
}

// ---------------- pass 1: bucket tokens by cluster --------------------
__global__ void bucket_tokens(const int* __restrict__ inp,
                              int* __restrict__ cnt,
                              int* __restrict__ list, int T) {
    int t = blockIdx.x * blockDim.x + threadIdx.x;
    if (t >= T) return;
    int v = inp[t];
    int c = (v < 20000) ? 0 : (v < 40000) ? 1 : (v < 200000) ? 2 : 3;
    int slot = atomicAdd(&cnt[c], 1);
    list[c * T + slot] = t;
}

// ---------------- pass 2: per-cluster WMMA GEMM -----------------------
// Block: 256 threads (8 wave32). Tile: 16 tokens x 128 output dims.
// Each wave: one 16x16 tile via V_WMMA_F32_16X16X4_F32, K-loop step 4.
template <int D, int CBASE>
__global__ __launch_bounds__(256)
void gemm_cluster(const int* __restrict__ inp,
                  const int* __restrict__ list,
                  const int* __restrict__ cnt,
                  const float* __restrict__ emb,
                  const float* __restrict__ proj,
                  float* __restrict__ out,
                  int T, int cluster) {
    constexpr int KT = (D < 256) ? D : 256;       // K chunk staged in LDS
    __shared__ float sA[16 * KT];                 // <= 16 KB
    __shared__ int   sTok[16];
    __shared__ int   sLoc[16];

    const int count = cnt[cluster];
    const int r0 = blockIdx.x * 16;
    if (r0 >= count) return;                      // uniform exit: EXEC stays full

    const int tid = threadIdx.x;
    if (tid < 16) {
        int r = r0 + tid;
        int tok = (r < count) ? list[cluster * T + r] : -1;
        sTok[tid] = tok;
        sLoc[tid] = (tok >= 0) ? (inp[tok] - CBASE) : 0;
    }
    __syncthreads();

    const int wave = tid >> 5;        // 0..7
    const int lane = tid & 31;
    const int half = lane >> 4;       // 0/1 -> K pair {2h, 2h+1}
    const int l16  = lane & 15;       // M index (A) / N index (B,C)
    const int p0   = blockIdx.y * 128 + wave * 16;

    // B fragment base: B[k][n] = proj[(p0+n)*D + k]; lane reads 2 consecutive K
    const float* projN = proj + (size_t)(p0 + l16) * D + half * 2;
    const float* aRow  = sA + l16 * KT + half * 2;

    v8f acc = {0.f, 0.f, 0.f, 0.f, 0.f, 0.f, 0.f, 0.f};

    for (int kb = 0; kb < D; kb += KT) {
        // cooperative gather of 16 rows x KT cols into LDS (zero pad invalid rows)
        for (int i = tid; i < 16 * KT / 4; i += 256) {
            int row = i / (KT / 4);
            int col = (i % (KT / 4)) * 4;
            v4f val = {0.f, 0.f, 0.f, 0.f};
            if (sTok[row] >= 0)
                val = *(const v4f*)(emb + (size_t)sLoc[row] * D + kb + col);
            *(v4f*)(sA + row * KT + col) = val;
        }
        __syncthreads();

        if (kb + KT < D)                          // prefetch next B chunk
            __builtin_prefetch(projN + kb + KT, 0, 0);

        #pragma unroll 8
        for (int k = 0; k < KT; k += 4) {
            v2f a = *(const v2f*)(aRow + k);              // ds_load_b64
            v2f b = *(const v2f*)(projN + kb + k);        // global_load_b64
            acc = __builtin_amdgcn_wmma_f32_16x16x4_f32(
                false, a, false, b, (short)0, acc, false, false);
        }
        __syncthreads();
    }

    // C layout: register v -> row M = 8*half + v, col N = l16
    #pragma unroll
    for (int v = 0; v < 8; ++v) {
        int m = half * 8 + v;
        if (r0 + m < count) {
            int tok = sTok[m];
            out[(size_t)tok * D_PROJ + p0 + l16] = acc[v] * EMB_SCALE;
        }
    }
}

extern "C" void kernel_launch(void* const* d_in, const int* in_sizes, int n_in,
                              void* d_out, int out_size, void* d_ws, size_t ws_size,
                              hipStream_t stream) {
    // setup_inputs order: inp, emb0, proj0, emb1, proj1, emb2, proj2, emb3, proj3
    const int*   inp   = (const int*)d_in[0];
    const float* emb0  = (const float*)d_in[1];
    const float* proj0 = (const float*)d_in[2];
    const float* emb1  = (const float*)d_in[3];
    const float* proj1 = (const float*)d_in[4];
    const float* emb2  = (const float*)d_in[5];
    const float* proj2 = (const float*)d_in[6];
    const float* emb3  = (const float*)d_in[7];
    const float* proj3 = (const float*)d_in[8];
    float* out = (float*)d_out;
    const int T = in_sizes[0];                    // 16384

    int* cnt  = (int*)d_ws;                      // 4 counters
    int* list = cnt + 16;                        // 4 lists of T token ids

    init_counters<<<1, 32, 0, stream>>>(cnt);
    bucket_tokens<<<(T + 255) / 256, 256, 0, stream>>>(inp, cnt, list, T);

    dim3 grid((T + 15) / 16, D_PROJ / 128);      // most row-blocks early-exit on count
    dim3 block(256);
    gemm_cluster<1024,      0><<<grid, block, 0, stream>>>(inp, list, cnt, emb0, proj0, out, T, 0);
    gemm_cluster< 256,  20000><<<grid, block, 0, stream>>>(inp, list, cnt, emb1, proj1, out, T, 1);
    gemm_cluster<  64,  40000><<<grid, block, 0, stream>>>(inp, list, cnt, emb2, proj2, out, T, 2);
    gemm_cluster<  16, 200000><<<grid, block, 0, stream>>>(inp, list, cnt, emb3, proj3, out, T, 3);
}